// ImageCube_17222818857178
// MI455X (gfx1250) — compile-verified
//
#include <hip/hip_runtime.h>
#include <hip/hip_bf16.h>

// ---------------------------------------------------------------------------
// ImageCube fwd: vis = 0.005^2 * rfft2(cube*corrfun), then per-channel sparse
// gather-dot.  FFT = Cooley-Tukey 1024 = 16 x 64 with both sub-DFTs executed
// as dense matmuls on V_WMMA_F32_16X16X4_F32 (wave32 fragment layouts per
// CDNA5 ISA 7.12.2).  Memory-bound problem (~1.5-2 GB -> ~80us @ 23.3 TB/s),
// so all global phases are coalesced/32B-chunked.
// ---------------------------------------------------------------------------

typedef float v2f __attribute__((ext_vector_type(2)));
typedef float v8f __attribute__((ext_vector_type(8)));

#define NCHAN 32
#define NPIX  1024
#define NCOL  513                       // npix/2 + 1
#define NVIS  50000
#define KNNZ  36
#define MFLAT (NPIX * NCOL)             // 525312
#define SCALE 2.5e-5f                   // (0.005*arcsec)^2 / arcsec^2 ... ^2 -> 0.005^2

static __device__ __forceinline__ v8f wmma4(v2f a, v2f b, v8f c) {
  // D(16x16,f32) = A(16x4,f32) * B(4x16,f32) + C   (8-arg f32/f16/bf16 form)
  return __builtin_amdgcn_wmma_f32_16x16x4_f32(false, a, false, b, (short)0, c,
                                               false, false);
}

// In-place complex 1024-point DFT on LDS arrays re[1024], im[1024].
// Executed cooperatively by ONE wave32 (EXEC all ones; no divergence inside).
//   n = 64*n1 + n2,  k = k1 + 16*k2,  n1,k1 in [0,16), n2,k2 in [0,64)
//   X[k1+16k2] = sum_n2 W64^(n2 k2) * W1024^(n2 k1) * sum_n1 W16^(n1 k1) x[64n1+n2]
static __device__ __forceinline__ void wave_fft1024(float* re, float* im) {
  const int lane = threadIdx.x & 31;
  const int l15  = lane & 15;
  const int hi   = lane >> 4;                   // 0 or 1 (half-wave)
  const float TWO_PI = 6.28318530717958647692f;
  const float C16 = TWO_PI / 16.0f, C64 = TWO_PI / 64.0f,
              C1024 = TWO_PI / 1024.0f;

  // DFT16 A-fragments (16x4 chunks): lane<16 holds (M=lane, K=0/1),
  // lane>=16 holds (M=lane-16, K=2/3); v2f component e = VGPR e.
  v2f aC[4], aS[4], aSn[4];
#pragma unroll
  for (int kk = 0; kk < 4; ++kk) {
    int kb = 4 * kk + (hi << 1);
#pragma unroll
    for (int e = 0; e < 2; ++e) {
      float s, c;
      __sincosf(C16 * (float)((l15 * (kb + e)) & 15), &s, &c);
      aC[kk][e] = c; aS[kk][e] = s; aSn[kk][e] = -s;
    }
  }

  // ---- Stage 1: Y[k1][n2] = DFT16 * x, then twiddle W1024^(n2 k1); the tile
  // over n2 reads/writes the same LDS address set {r*64+n2}, so it is in-place.
#pragma unroll
  for (int t = 0; t < 4; ++t) {
    const int n2 = t * 16 + l15;
    v2f bre[4], bim[4];                         // B 4x16: K=0/2 by half-wave
#pragma unroll
    for (int kk = 0; kk < 4; ++kk) {
      int n1 = 4 * kk + (hi << 1);
      bre[kk][0] = re[n1 * 64 + n2];  bre[kk][1] = re[(n1 + 1) * 64 + n2];
      bim[kk][0] = im[n1 * 64 + n2];  bim[kk][1] = im[(n1 + 1) * 64 + n2];
    }
    v8f yr = {0, 0, 0, 0, 0, 0, 0, 0}, yi = {0, 0, 0, 0, 0, 0, 0, 0};
#pragma unroll
    for (int kk = 0; kk < 4; ++kk) {            // W16 = cos - i sin
      yr = wmma4(aC[kk],  bre[kk], yr);         // Yre = c*xre + s*xim
      yr = wmma4(aS[kk],  bim[kk], yr);
      yi = wmma4(aC[kk],  bim[kk], yi);         // Yim = c*xim - s*xre
      yi = wmma4(aSn[kk], bre[kk], yi);
    }
    // D element (M,N) = (k1,n2): apply W1024^(n2*k1) per lane, store in place.
#pragma unroll
    for (int v = 0; v < 8; ++v) {
      int k1 = v + (hi << 3);
      float s, c;
      __sincosf(C1024 * (float)((n2 * k1) & 1023), &s, &c);
      float r = yr[v], q = yi[v];
      re[k1 * 64 + n2] = c * r + s * q;
      im[k1 * 64 + n2] = c * q - s * r;
    }
  }

  // ---- Stage 2: X[k1+16k2] = Ytw(16x64) * W64(64x64); data is operand A.
  v8f zr[4] = {}, zi[4] = {};
  for (int tt = 0; tt < 4; ++tt) {
    const int k2 = tt * 16 + l15;               // B operand N index
    for (int cc = 0; cc < 16; ++cc) {
      const int nb = 4 * cc + (hi << 1);        // K base for this half-wave
      v2f ar, ai, bc, bs, bn;
      ar[0] = re[l15 * 64 + nb];  ar[1] = re[l15 * 64 + nb + 1];
      ai[0] = im[l15 * 64 + nb];  ai[1] = im[l15 * 64 + nb + 1];
#pragma unroll
      for (int e = 0; e < 2; ++e) {
        float s, c;
        __sincosf(C64 * (float)(((nb + e) * k2) & 63), &s, &c);
        bc[e] = c; bs[e] = s; bn[e] = -s;
      }
      zr[tt] = wmma4(ar, bc, zr[tt]);           // Zre = Are*c + Aim*s
      zr[tt] = wmma4(ai, bs, zr[tt]);
      zi[tt] = wmma4(ai, bc, zi[tt]);           // Zim = Aim*c - Are*s
      zi[tt] = wmma4(ar, bn, zi[tt]);
    }
  }
#pragma unroll
  for (int tt = 0; tt < 4; ++tt)
#pragma unroll
    for (int v = 0; v < 8; ++v) {
      int k1  = v + (hi << 3);
      int idx = k1 + 16 * (tt * 16 + l15);      // k = k1 + 16*k2
      re[idx] = zr[tt][v];
      im[idx] = zi[tt][v];
    }
}

// Pass 1: per-row real FFT. 8 waves/block = 8 rows; coalesced loads; the 513
// kept bins are transpose-stored to [c][col][row] in 8-row (32B) chunks.
__global__ __launch_bounds__(256) void fft_rows(const float* __restrict__ cube,
                                                const float* __restrict__ corr,
                                                float* __restrict__ REt,
                                                float* __restrict__ IMt) {
  __shared__ float sre[8 * 1024];
  __shared__ float sim[8 * 1024];
  const int t = threadIdx.x, w = t >> 5, lane = t & 31;
  const int c = blockIdx.y, row0 = blockIdx.x * 8, row = row0 + w;
  const float* src = cube + ((size_t)c * NPIX + row) * NPIX;
  const float* cf  = corr + (size_t)row * NPIX;
#pragma unroll
  for (int i = 0; i < 32; ++i) {
    int n = i * 32 + lane;
    sre[w * 1024 + n] = src[n] * cf[n] * SCALE;
    sim[w * 1024 + n] = 0.0f;
  }
  wave_fft1024(sre + w * 1024, sim + w * 1024);   // wave-private region
  __syncthreads();
  const int r8 = t & 7, cb = t >> 3;              // 8 rows x 32 col-slots
  for (int i = 0; i < 17; ++i) {
    int col = cb + 32 * i;
    if (col < NCOL) {
      size_t o = ((size_t)c * NCOL + col) * NPIX + row0 + r8;
      REt[o] = sre[r8 * 1024 + col];
      IMt[o] = sim[r8 * 1024 + col];
    }
  }
}

// Pass 2: per-column complex FFT. Contiguous 4KB column loads; results
// scattered into vr/vi[c][k*513+col] in 8-col (32B) chunks.
__global__ __launch_bounds__(256) void fft_cols(const float* __restrict__ REt,
                                                const float* __restrict__ IMt,
                                                float* __restrict__ VR,
                                                float* __restrict__ VI) {
  __shared__ float sre[8 * 1024];
  __shared__ float sim[8 * 1024];
  const int t = threadIdx.x, w = t >> 5, lane = t & 31;
  const int c = blockIdx.y, col0 = blockIdx.x * 8, col = col0 + w;
  if (col < NCOL) {                               // wave-uniform branch
    const float* pr = REt + ((size_t)c * NCOL + col) * NPIX;
    const float* pi = IMt + ((size_t)c * NCOL + col) * NPIX;
#pragma unroll
    for (int i = 0; i < 32; ++i) {
      int n = i * 32 + lane;
      sre[w * 1024 + n] = pr[n];
      sim[w * 1024 + n] = pi[n];
    }
  } else {
    for (int i = 0; i < 32; ++i) {
      int n = i * 32 + lane;
      sre[w * 1024 + n] = 0.f; sim[w * 1024 + n] = 0.f;
    }
  }
  wave_fft1024(sre + w * 1024, sim + w * 1024);
  __syncthreads();
  const int c8 = t & 7, kb = t >> 3;
  const int colS = col0 + c8;
  if (colS < NCOL) {
    float* vr = VR + (size_t)c * MFLAT;
    float* vi = VI + (size_t)c * MFLAT;
#pragma unroll 4
    for (int i = 0; i < 32; ++i) {
      int k = kb + 32 * i;
      vr[(size_t)k * NCOL + colS] = sre[c8 * 1024 + k];
      vi[(size_t)k * NCOL + colS] = sim[c8 * 1024 + k];
    }
  }
}

// Sparse interpolation: one thread per (c,j); idx/val stream, gathers hit the
// 67MB vr/vi tables resident in the 192MB L2.
__global__ __launch_bounds__(256) void gather_interp(
    const int* __restrict__ reI, const float* __restrict__ reV,
    const int* __restrict__ imI, const float* __restrict__ imV,
    const float* __restrict__ VR, const float* __restrict__ VI,
    float* __restrict__ out) {
  int gid = blockIdx.x * 256 + threadIdx.x;
  if (gid >= NCHAN * NVIS) return;
  int c = gid / NVIS;
  const float* vr = VR + (size_t)c * MFLAT;
  const float* vi = VI + (size_t)c * MFLAT;
  size_t base = (size_t)gid * KNNZ;
  float accr = 0.f, acci = 0.f;
#pragma unroll 4
  for (int k = 0; k < KNNZ; ++k) {
    accr += reV[base + k] * vr[reI[base + k]];
    acci += imV[base + k] * vi[imI[base + k]];
  }
  out[gid] = accr;                      // re block
  out[NCHAN * NVIS + gid] = acci;       // im block
}

extern "C" void kernel_launch(void* const* d_in, const int* in_sizes, int n_in,
                              void* d_out, int out_size, void* d_ws,
                              size_t ws_size, hipStream_t stream) {
  const float* cube = (const float*)d_in[0];
  const float* corr = (const float*)d_in[1];
  const int*   reI  = (const int*)d_in[2];
  const float* reV  = (const float*)d_in[3];
  const int*   imI  = (const int*)d_in[4];
  const float* imV  = (const float*)d_in[5];
  float* out = (float*)d_out;

  float* w = (float*)d_ws;
  const size_t P = (size_t)NCHAN * MFLAT;   // 16,809,984 floats per plane
  float* REt = w;                            // [c][col][row]  (pass1 out)
  float* IMt = w + P;
  float* VR  = w + 2 * P;                    // [c][k*513+col] (pass2 out)
  float* VI  = w + 3 * P;

  dim3 blk(256);
  fft_rows<<<dim3(NPIX / 8, NCHAN), blk, 0, stream>>>(cube, corr, REt, IMt);
  fft_cols<<<dim3((NCOL + 7) / 8, NCHAN), blk, 0, stream>>>(REt, IMt, VR, VI);
  gather_interp<<<dim3((NCHAN * NVIS + 255) / 256), blk, 0, stream>>>(
      reI, reV, imI, imV, VR, VI, out);
}